// EntityEncoder_69707319214162
// MI455X (gfx1250) — compile-verified
//
#include <hip/hip_runtime.h>
#include <math.h>

// ---------------- constants from the reference ----------------
#define ES 256
#define ES4 64      // ES / 4 (float4 granularity)
#define NF 42
#define NB 8192
#define NN 12

// feature indices
#define SPECIES 0
#define ABILITY 1
#define ITEM 2
#define ITEM_EFFECT 3
#define GENDER 4
#define STATUS 5
#define BEING_CALLED_BACK 6
#define TRAPPED 7
#define NEWLY_SWITCHED 8
#define TOXIC_TURNS 9
#define SLEEP_TURNS 10
#define FAINTED 11
#define ACTIVE 12
#define SIDE 13
#define LEVEL 14
#define HP 15
#define MAXHP 16
#define BOOST0 17
#define VOL0 24
#define MOVEID0 33
#define MOVEPP0 37

typedef float v2f __attribute__((ext_vector_type(2)));
typedef float v8f __attribute__((ext_vector_type(8)));

__device__ __forceinline__ void add4(float4& a, const float4 b) {
  a.x += b.x; a.y += b.y; a.z += b.z; a.w += b.w;
}
__device__ __forceinline__ void fma4(float4& a, float s, const float4 b) {
  a.x += s * b.x; a.y += s * b.y; a.z += s * b.z; a.w += s * b.w;
}
// row-gather of a float4 from a table of 256-float rows
__device__ __forceinline__ float4 row4(const float* __restrict__ base, size_t row, int c4) {
  return ((const float4*)base)[row * ES4 + c4];
}

// ---------------------------------------------------------------
// Kernel 1: P[npad x 256] = E[nrows x 256] @ W[256 x 256]
// One wave (32 threads) per 16x16 output tile, f32 WMMA 16x16x4.
// A layout (16x4 f32): lanes 0-15 -> M=lane, VGPR{0,1}=K{0,1};
//                      lanes 16-31 -> M=lane-16, VGPR{0,1}=K{2,3}.
// B layout mirrors A with N in place of M (B loaded as K x N).
// C/D layout: VGPR r, lanes 0-15 -> M=r, lanes 16-31 -> M=r+8; N=lane&15.
// P is padded to the grid's row coverage, so stores are unguarded.
// ---------------------------------------------------------------
__global__ __launch_bounds__(32) void proj_gemm_wmma(
    const float* __restrict__ E, const float* __restrict__ W,
    float* __restrict__ P, int nrows) {
  const int lane = threadIdx.x;
  const int tm = blockIdx.x >> 4;       // row tile
  const int tn = blockIdx.x & 15;       // col tile (256/16 = 16)
  const int row0 = tm * 16;
  const int col0 = tn * 16;

  const int mrow = row0 + (lane & 15);
  const int arow = (mrow < nrows) ? mrow : 0;      // clamp for safe addressing
  const int khalf = (lane >> 4) << 1;              // 0 for lanes 0-15, 2 for lanes 16-31
  const float* Abase = E + (size_t)arow * ES + khalf;
  const float* Bbase = W + col0 + (lane & 15);

  v8f c = {};
  for (int k0 = 0; k0 < ES; k0 += 4) {
    v2f a, b;
    a.x = Abase[k0];
    a.y = Abase[k0 + 1];
    const int kb = k0 + khalf;
    b.x = Bbase[(size_t)kb * ES];
    b.y = Bbase[(size_t)(kb + 1) * ES];
    c = __builtin_amdgcn_wmma_f32_16x16x4_f32(
        /*neg_a=*/false, a, /*neg_b=*/false, b,
        /*c_mod=*/(short)0, c, /*reuse_a=*/false, /*reuse_b=*/false);
  }

  const int colg = col0 + (lane & 15);
  const size_t mbase = (size_t)(row0 + ((lane >> 4) << 3));
#pragma unroll
  for (int r = 0; r < 8; ++r)
    P[(mbase + r) * ES + colg] = c[r];   // padded P: no guard needed
}

// ---------------------------------------------------------------
// Kernel 2: fold all bias vectors into one (moves_b counted 4x).
// ---------------------------------------------------------------
__global__ __launch_bounds__(256) void bias_fold(
    const float* hp_b, const float* level_b, const float* active_b,
    const float* onehot_b, const float* boosts_b, const float* volatiles_b,
    const float* item_b, const float* side_b, const float* moves_b,
    float* __restrict__ out) {
  const int d = threadIdx.x;
  out[d] = hp_b[d] + level_b[d] + active_b[d] + onehot_b[d] + boosts_b[d] +
           volatiles_b[d] + item_b[d] + side_b[d] + 4.0f * moves_b[d];
}

// ---------------------------------------------------------------
// Kernel 3: main encoder — 4 entities per 256-thread block, 64
// threads per entity, each thread owns 4 channels (float4 / b128
// gathers: 512 B per wave per table row). Threads 0-63 -> entity 0
// etc., so every wave is entity-uniform and all branches are
// wave-uniform.
// ---------------------------------------------------------------
__global__ __launch_bounds__(256) void entity_encode(
    const int* __restrict__ ents,
    const float* __restrict__ embed_species,
    const float* __restrict__ embed_ability,
    const float* __restrict__ item_proj,   // 512 x 256, precomputed
    const float* __restrict__ item_W,      // 272 x 256 (rows 256.. = item_effect)
    const float* __restrict__ moves_proj,  // 928 x 256, precomputed
    const float* __restrict__ moves_W,     // 262 x 256 (rows 256.. = pp bits)
    const float* __restrict__ hp_W,        // 10 x 256
    const float* __restrict__ level_W,     // 7 x 256
    const float* __restrict__ active_W,    // 2 x 256
    const float* __restrict__ onehot_W,    // 32 x 256
    const float* __restrict__ boosts_W,    // 7 x 256
    const float* __restrict__ volatiles_W, // 105 x 256
    const float* __restrict__ side_W,      // 2 x 256
    const float* __restrict__ bias,        // 256 (prefolded)
    float* __restrict__ out) {
  __shared__ int es[4 * NF];
  const int* ep = ents + (size_t)blockIdx.x * (4 * NF);
  if (threadIdx.x < 4 * NF) es[threadIdx.x] = ep[threadIdx.x];
  __syncthreads();

  const int el = threadIdx.x >> 6;         // entity within block (wave-uniform)
  const int c4 = threadIdx.x & 63;         // float4 channel group
  const int* e = es + el * NF;

  float4 acc = ((const float4*)bias)[c4];

  // embedding gathers
  add4(acc, row4(embed_species, (size_t)e[SPECIES], c4));
  add4(acc, row4(embed_ability, (size_t)e[ABILITY], c4));
  add4(acc, row4(item_proj, (size_t)e[ITEM], c4));
  { unsigned v = (unsigned)e[ITEM_EFFECT];
    if (v < 16u) add4(acc, row4(item_W, 256 + v, c4)); }

  // hp token bits (10) + hp_ratio * onehot_W[0]
  {
    const float hp = (float)e[HP];
    const int mh = (e[MAXHP] < 1) ? 1 : e[MAXHP];
    float ratio = hp / (float)mh;
    ratio = ratio < 0.0f ? 0.0f : (ratio > 1.0f ? 1.0f : ratio);
    const int token = (int)floorf(1023.0f * ratio);
#pragma unroll
    for (int b = 0; b < 10; ++b)
      if ((token >> b) & 1) add4(acc, row4(hp_W, b, c4));
    fma4(acc, ratio, row4(onehot_W, 0, c4));   // boolean_code[0]
  }

  // level bits (7)
  {
    const int lvl = e[LEVEL];
#pragma unroll
    for (int b = 0; b < 7; ++b)
      if ((lvl >> b) & 1) add4(acc, row4(level_W, b, c4));
  }

  // one-hot selections (jax one_hot: out-of-range -> zero vector)
  { unsigned v = (unsigned)e[ACTIVE]; if (v < 2u) add4(acc, row4(active_W, v, c4)); }
  { unsigned v = (unsigned)e[SIDE];   if (v < 2u) add4(acc, row4(side_W, v, c4)); }
  { unsigned v = (unsigned)e[GENDER];            if (v < 3u) add4(acc, row4(onehot_W, 1 + v, c4)); }
  { unsigned v = (unsigned)e[STATUS];            if (v < 8u) add4(acc, row4(onehot_W, 4 + v, c4)); }
  { unsigned v = (unsigned)e[BEING_CALLED_BACK]; if (v < 2u) add4(acc, row4(onehot_W, 12 + v, c4)); }
  { unsigned v = (unsigned)e[TRAPPED];           if (v < 2u) add4(acc, row4(onehot_W, 14 + v, c4)); }
  { unsigned v = (unsigned)e[NEWLY_SWITCHED];    if (v < 2u) add4(acc, row4(onehot_W, 16 + v, c4)); }
  { unsigned v = (unsigned)e[TOXIC_TURNS];       if (v < 8u) add4(acc, row4(onehot_W, 18 + v, c4)); }
  { unsigned v = (unsigned)e[SLEEP_TURNS];       if (v < 4u) add4(acc, row4(onehot_W, 26 + v, c4)); }
  { unsigned v = (unsigned)e[FAINTED];           if (v < 2u) add4(acc, row4(onehot_W, 30 + v, c4)); }

  // boosts: (value / 2) * row
#pragma unroll
  for (int i = 0; i < 7; ++i)
    fma4(acc, 0.5f * (float)e[BOOST0 + i], row4(boosts_W, i, c4));

  // volatiles: NOTE reference uses (vol & j) where j is the *position*
  // 0..15 (not 1<<j) — replicate exactly. 144 flags truncated to 105.
  for (int k = 0; k < 105; ++k) {
    const int vi = e[VOL0 + (k >> 4)];
    if (vi & (k & 15)) add4(acc, row4(volatiles_W, k, c4));
  }

  // moves: projected embedding gather + pp bit rows (6 bits, world 64)
#pragma unroll
  for (int m = 0; m < 4; ++m) {
    add4(acc, row4(moves_proj, (size_t)e[MOVEID0 + m], c4));
    const int pp = e[MOVEPP0 + m];
#pragma unroll
    for (int b = 0; b < 6; ++b)
      if ((pp >> b) & 1) add4(acc, row4(moves_W, 256 + b, c4));
  }

  const size_t gent = (size_t)blockIdx.x * 4 + el;
  ((float4*)out)[gent * ES4 + c4] = acc;
}

// ---------------------------------------------------------------
extern "C" void kernel_launch(void* const* d_in, const int* in_sizes, int n_in,
                              void* d_out, int out_size, void* d_ws, size_t ws_size,
                              hipStream_t stream) {
  const int*   entities      = (const int*)  d_in[0];
  const float* embed_species = (const float*)d_in[1];
  const float* embed_ability = (const float*)d_in[2];
  const float* embed_item    = (const float*)d_in[3];
  const float* embed_moves   = (const float*)d_in[4];
  const float* hp_W        = (const float*)d_in[5];
  const float* hp_b        = (const float*)d_in[6];
  const float* level_W     = (const float*)d_in[7];
  const float* level_b     = (const float*)d_in[8];
  const float* active_W    = (const float*)d_in[9];
  const float* active_b    = (const float*)d_in[10];
  const float* onehot_W    = (const float*)d_in[11];
  const float* onehot_b    = (const float*)d_in[12];
  const float* boosts_W    = (const float*)d_in[13];
  const float* boosts_b    = (const float*)d_in[14];
  const float* volatiles_W = (const float*)d_in[15];
  const float* volatiles_b = (const float*)d_in[16];
  const float* item_W      = (const float*)d_in[17];
  const float* item_b      = (const float*)d_in[18];
  const float* side_W      = (const float*)d_in[19];
  const float* side_b      = (const float*)d_in[20];
  const float* moves_W     = (const float*)d_in[21];
  const float* moves_b     = (const float*)d_in[22];

  // workspace layout (~1.5 MB): padded projected tables + folded bias
  float* item_proj  = (float*)d_ws;              // 512 x 256
  float* moves_proj = item_proj + 512 * ES;      // 928 x 256
  float* bias       = moves_proj + 928 * ES;     // 256

  // Precompute projected gather tables with f32 WMMA (exact f32 math):
  //   item_proj  = embed_item  @ item_W[0:256,:]   (500x256x256)
  //   moves_proj = embed_moves @ moves_W[0:256,:]  (920x256x256)
  proj_gemm_wmma<<<32 * 16, 32, 0, stream>>>(embed_item,  item_W,  item_proj,  500);
  proj_gemm_wmma<<<58 * 16, 32, 0, stream>>>(embed_moves, moves_W, moves_proj, 920);

  bias_fold<<<1, 256, 0, stream>>>(hp_b, level_b, active_b, onehot_b, boosts_b,
                                   volatiles_b, item_b, side_b, moves_b, bias);

  // Main bandwidth-bound pass: 4 entities/block, float4 (b128) gathers.
  entity_encode<<<(NB * NN) / 4, 256, 0, stream>>>(
      entities, embed_species, embed_ability,
      item_proj, item_W, moves_proj, moves_W,
      hp_W, level_W, active_W, onehot_W, boosts_W, volatiles_W, side_W,
      bias, (float*)d_out);
}